// CrossAttentionBlock_73452530696666
// MI455X (gfx1250) — compile-verified
//
#include <hip/hip_runtime.h>

typedef __attribute__((ext_vector_type(2))) float v2f;
typedef __attribute__((ext_vector_type(4))) float f4;
typedef __attribute__((ext_vector_type(8))) float v8f;

#define B_DIM 16
#define C_DIM 512
#define T_DIM 768
#define S_DIM 4096
#define LN_EPS 1e-5f

// ---------------------------------------------------------------------------
// Kernel A: V(16x512) = E(16x768) @ Wv^T   using V_WMMA_F32_16X16X4_F32
// One wave per block; block n owns output columns [n*16, n*16+16).
// A 16x4 layout: lanes 0-15 -> K = k+0,k+1 ; lanes 16-31 -> K = k+2,k+3
// B 4x16 layout mirrored (lane&15 = column, K split by lane half).
// D 16x16 layout: VGPR r -> M = r (lanes 0-15) / r+8 (lanes 16-31), N = lane&15
// ---------------------------------------------------------------------------
__global__ __launch_bounds__(32) void proj_v_kernel(const float* __restrict__ E,
                                                    const float* __restrict__ Wv,
                                                    float* __restrict__ V) {
    const int lane  = threadIdx.x;
    const int row   = lane & 15;
    const int k0    = (lane >> 4) << 1;          // 0 or 2
    const int ntile = blockIdx.x * 16;
    const int ncol  = ntile + row;

    const float* arow = E  + row  * T_DIM + k0;
    const float* brow = Wv + ncol * T_DIM + k0;  // B[k,n] = Wv[n,k]

    v8f acc = {};
    for (int k = 0; k < T_DIM; k += 4) {
        v2f a = *(const v2f*)(arow + k);
        v2f b = *(const v2f*)(brow + k);
        acc = __builtin_amdgcn_wmma_f32_16x16x4_f32(false, a, false, b,
                                                    (short)0, acc, false, false);
    }

    const int mrow = (lane >> 4) << 3;           // 0 or 8
    #pragma unroll
    for (int r = 0; r < 8; ++r) {
        V[(size_t)(mrow + r) * C_DIM + ncol] = acc[r];
    }
}

// ---------------------------------------------------------------------------
// Kernel B: O(16x512) = V(16x512) @ Wo^T + bo
// ---------------------------------------------------------------------------
__global__ __launch_bounds__(32) void proj_o_kernel(const float* __restrict__ V,
                                                    const float* __restrict__ Wo,
                                                    const float* __restrict__ bo,
                                                    float* __restrict__ O) {
    const int lane  = threadIdx.x;
    const int row   = lane & 15;
    const int k0    = (lane >> 4) << 1;
    const int ntile = blockIdx.x * 16;
    const int ncol  = ntile + row;

    const float* arow = V  + row  * C_DIM + k0;
    const float* brow = Wo + ncol * C_DIM + k0;  // B[k,n] = Wo[n,k]

    v8f acc = {};
    for (int k = 0; k < C_DIM; k += 4) {
        v2f a = *(const v2f*)(arow + k);
        v2f b = *(const v2f*)(brow + k);
        acc = __builtin_amdgcn_wmma_f32_16x16x4_f32(false, a, false, b,
                                                    (short)0, acc, false, false);
    }

    const float bias = bo[ncol];
    const int mrow = (lane >> 4) << 3;
    #pragma unroll
    for (int r = 0; r < 8; ++r) {
        O[(size_t)(mrow + r) * C_DIM + ncol] = acc[r] + bias;
    }
}

// ---------------------------------------------------------------------------
// Kernel C: y[b,c,s] = LN_c( x[b,c,s] + O[b,c] ) * gamma[c] + beta[c]
// 128 threads/block, each thread owns 4 consecutive s (float4 coalesced).
// Block covers 512 s positions of one batch; grid = 16 * 8 = 128 blocks.
// Pass 1: moments. Pass 2: normalize (re-read hits L2), NT stores out.
// ---------------------------------------------------------------------------
__global__ __launch_bounds__(128) void fused_ln_kernel(const float* __restrict__ x,
                                                       const float* __restrict__ O,
                                                       const float* __restrict__ gamma,
                                                       const float* __restrict__ beta,
                                                       float* __restrict__ y) {
    __shared__ float o_s[C_DIM];
    __shared__ float g_s[C_DIM];
    __shared__ float b_s[C_DIM];

    const int tid   = threadIdx.x;
    const int b     = blockIdx.x >> 3;
    const int stile = (blockIdx.x & 7) << 9;     // * 512

    for (int i = tid; i < C_DIM; i += 128) {
        o_s[i] = O[b * C_DIM + i];
        g_s[i] = gamma[i];
        b_s[i] = beta[i];
    }
    __syncthreads();

    const size_t base = (size_t)b * C_DIM * S_DIM + (size_t)stile + ((size_t)tid << 2);
    const float* xb = x + base;
    float*       yb = y + base;

    float s0 = 0.f, s1 = 0.f, s2 = 0.f, s3 = 0.f;
    float q0 = 0.f, q1 = 0.f, q2 = 0.f, q3 = 0.f;

    #pragma unroll 8
    for (int c = 0; c < C_DIM; ++c) {
        f4 v = *(const f4*)(xb + (size_t)c * S_DIM);
        const float oc = o_s[c];
        const float h0 = v.x + oc, h1 = v.y + oc, h2 = v.z + oc, h3 = v.w + oc;
        s0 += h0; q0 += h0 * h0;
        s1 += h1; q1 += h1 * h1;
        s2 += h2; q2 += h2 * h2;
        s3 += h3; q3 += h3 * h3;
    }

    const float inv = 1.0f / (float)C_DIM;
    const float mu0 = s0 * inv, mu1 = s1 * inv, mu2 = s2 * inv, mu3 = s3 * inv;
    const float r0 = rsqrtf(q0 * inv - mu0 * mu0 + LN_EPS);
    const float r1 = rsqrtf(q1 * inv - mu1 * mu1 + LN_EPS);
    const float r2 = rsqrtf(q2 * inv - mu2 * mu2 + LN_EPS);
    const float r3 = rsqrtf(q3 * inv - mu3 * mu3 + LN_EPS);

    #pragma unroll 8
    for (int c = 0; c < C_DIM; ++c) {
        f4 v = *(const f4*)(xb + (size_t)c * S_DIM);
        const float oc = o_s[c];
        const float g  = g_s[c];
        const float bb = b_s[c];
        f4 out;
        out.x = (v.x + oc - mu0) * r0 * g + bb;
        out.y = (v.y + oc - mu1) * r1 * g + bb;
        out.z = (v.z + oc - mu2) * r2 * g + bb;
        out.w = (v.w + oc - mu3) * r3 * g + bb;
        __builtin_nontemporal_store(out, (f4*)(yb + (size_t)c * S_DIM));
    }
}

// ---------------------------------------------------------------------------
extern "C" void kernel_launch(void* const* d_in, const int* in_sizes, int n_in,
                              void* d_out, int out_size, void* d_ws, size_t ws_size,
                              hipStream_t stream) {
    const float* x        = (const float*)d_in[0];
    const float* text_emb = (const float*)d_in[1];
    // d_in[2] = Wq, d_in[3] = Wk : provably unused — softmax over a single
    // key token is identically 1.0, so q/k never influence the output.
    const float* Wv    = (const float*)d_in[4];
    const float* Wo    = (const float*)d_in[5];
    const float* bo    = (const float*)d_in[6];
    const float* gamma = (const float*)d_in[7];
    const float* beta  = (const float*)d_in[8];
    float* out = (float*)d_out;

    float* V = (float*)d_ws;            // 16*512 floats
    float* O = V + B_DIM * C_DIM;       // 16*512 floats

    proj_v_kernel<<<C_DIM / 16, 32, 0, stream>>>(text_emb, Wv, V);
    proj_o_kernel<<<C_DIM / 16, 32, 0, stream>>>(V, Wo, bo, O);
    fused_ln_kernel<<<B_DIM * (S_DIM / 512), 128, 0, stream>>>(x, O, gamma, beta, out);
}